// AttentionConv_self_58548994179158
// MI455X (gfx1250) — compile-verified
//
#include <hip/hip_runtime.h>

typedef float v2f __attribute__((ext_vector_type(2)));
typedef float v8f __attribute__((ext_vector_type(8)));
typedef unsigned int v4u __attribute__((ext_vector_type(4)));
typedef int v4i __attribute__((ext_vector_type(4)));
typedef int v8i __attribute__((ext_vector_type(8)));

#define CIN   64
#define COUT  64
#define HWDIM 64
#define NPIX  4096       // 64*64 pixels per (batch, channel) plane
#define KWIN  7
#define PADW  3

// ---------------------------------------------------------------------------
// Tensor Data Mover: DMA a contiguous row of `nelem` fp32 values from global
// memory into LDS at byte offset `lds_off`. Descriptor per CDNA5 ISA §8:
//   Group0: count=1 | lds_addr | global_addr[31:0] | global_addr[56:32],type=2
//   Group1: data_size=2 (4B); tensor_dim0 = tile_dim0 = nelem; tensor_dim1 =
//           tile_dim1 = 1; stride0 = stride1 = nelem. Groups 2/3 unused (2-D).
// Tracked by TENSORcnt (s_wait_tensorcnt).
// ---------------------------------------------------------------------------
__device__ __forceinline__ void tdm_load_row(const void* gptr, unsigned lds_off,
                                             unsigned nelem) {
  const unsigned long long ga = (unsigned long long)(uintptr_t)gptr;
  v4u g0;
  g0[0] = 1u;                                            // count=1 (valid)
  g0[1] = lds_off;                                       // LDS byte address
  g0[2] = (unsigned)(ga & 0xffffffffu);                  // global_addr lo
  g0[3] = (unsigned)((ga >> 32) & 0x1ffffffu) | (2u << 30);  // hi | type=2
  v8i g1;
  g1[0] = 0x00020000;          // wg_mask=0, data_size=2 (4B), no iter/pad
  g1[1] = (int)(nelem << 16);  // [31:16] = tensor_dim0[15:0]
  g1[2] = (int)(1u << 16);     // [15:0]=dim0 hi=0, [31:16]=tensor_dim1=1
  g1[3] = (int)(nelem << 16);  // [15:0]=dim1 hi=0, [31:16]=tile_dim0
  g1[4] = 1;                   // tile_dim1=1, tile_dim2=0
  g1[5] = (int)nelem;          // tensor_dim0_stride[31:0]
  g1[6] = (int)(nelem << 16);  // stride0 hi=0, tensor_dim1_stride[15:0]
  g1[7] = 0;                   // tensor_dim1_stride hi
  v4i z4 = {0, 0, 0, 0};
#if defined(__clang_major__) && (__clang_major__ >= 23)
  v8i z8 = {0, 0, 0, 0, 0, 0, 0, 0};
  __builtin_amdgcn_tensor_load_to_lds(g0, g1, z4, z4, z8, 0);
#else
  __builtin_amdgcn_tensor_load_to_lds(g0, g1, z4, z4, 0);
#endif
}

// ---------------------------------------------------------------------------
// Stage 1: fused Q/K/V 1x1-conv projections as fp32 WMMA GEMMs.
//   D[o, p] = sum_c W[o][c] * X[c][p]   (per batch)
// Weights are TDM-staged into LDS once per workgroup; each wave then runs a
// 16-pixel tile x all 64 output channels x {q,k,v}:
//   16 K-steps x 4 o-tiles x 3 mats = 192 v_wmma_f32_16x16x4_f32 per wave,
// with A-operands served by ds_load_b64 from LDS.
// ---------------------------------------------------------------------------
__global__ __launch_bounds__(256) void qkv_wmma_kernel(
    const float* __restrict__ x,
    const float* __restrict__ wq,
    const float* __restrict__ wk,
    const float* __restrict__ wv,
    float* __restrict__ qo,
    float* __restrict__ ko,
    float* __restrict__ vo)
{
  __shared__ float wsm[3 * COUT * CIN];   // 48 KB: wq | wk | wv

  if ((threadIdx.x >> 5) == 0) {          // wave 0 drives the TDM
    const unsigned base = (unsigned)(uintptr_t)wsm;
    tdm_load_row(wq, base,                          COUT * CIN);
    tdm_load_row(wk, base + 4u * COUT * CIN,        COUT * CIN);
    tdm_load_row(wv, base + 8u * COUT * CIN,        COUT * CIN);
    __builtin_amdgcn_s_wait_tensorcnt(0);
  }
  __syncthreads();

  const int lane  = threadIdx.x & 31;
  const int wave  = (blockIdx.x * blockDim.x + threadIdx.x) >> 5;
  const int b     = wave >> 8;          // 256 pixel-tiles per batch image
  const int p0    = (wave & 255) << 4;  // 16-pixel tile base
  const int khalf = lane >> 4;          // which K-pair this lane half holds
  const int ln    = lane & 15;          // row (A) / column (B,D) index

  const float* xb  = x + (size_t)b * (CIN * NPIX);
  const float* lwq = wsm;
  const float* lwk = wsm + COUT * CIN;
  const float* lwv = wsm + 2 * COUT * CIN;

  v8f accQ[4], accK[4], accV[4];
#pragma unroll
  for (int t = 0; t < 4; ++t) { accQ[t] = {}; accK[t] = {}; accV[t] = {}; }

#pragma unroll
  for (int ks = 0; ks < CIN / 4; ++ks) {
    const int c = (ks << 2) + (khalf << 1);
    // B operand: X[c..c+1][p0+ln]  (channel stride NPIX)
    v2f bm;
    bm.x = xb[(size_t)c * NPIX + p0 + ln];
    bm.y = xb[(size_t)(c + 1) * NPIX + p0 + ln];
#pragma unroll
    for (int t = 0; t < 4; ++t) {
      const int row = (t << 4) + ln;   // output channel row of A
      v2f aq = *(const v2f*)(lwq + row * CIN + c);
      v2f ak = *(const v2f*)(lwk + row * CIN + c);
      v2f av = *(const v2f*)(lwv + row * CIN + c);
      accQ[t] = __builtin_amdgcn_wmma_f32_16x16x4_f32(
          false, aq, false, bm, (short)0, accQ[t], false, false);
      accK[t] = __builtin_amdgcn_wmma_f32_16x16x4_f32(
          false, ak, false, bm, (short)0, accK[t], false, false);
      accV[t] = __builtin_amdgcn_wmma_f32_16x16x4_f32(
          false, av, false, bm, (short)0, accV[t], false, false);
    }
  }

  // Scatter D: VGPR r holds row (r + 8*khalf) of the 16x16 tile, lane = col.
#pragma unroll
  for (int t = 0; t < 4; ++t) {
#pragma unroll
    for (int r = 0; r < 8; ++r) {
      const int o = (t << 4) + r + (khalf << 3);
      const size_t off = ((size_t)(b * COUT + o)) * NPIX + p0 + ln;
      qo[off] = accQ[t][r];
      ko[off] = accK[t][r];
      vo[off] = accV[t][r];
    }
  }
}

// ---------------------------------------------------------------------------
// Stage 2: per-channel 7x7 local attention. One thread per output element.
// Out-of-bounds window taps behave as zero-padded conv: k = rel only, v = 0.
// ---------------------------------------------------------------------------
__global__ __launch_bounds__(256) void attn_kernel(
    const float* __restrict__ q,
    const float* __restrict__ kf,
    const float* __restrict__ vf,
    const float* __restrict__ rel_h,   // [32][7]
    const float* __restrict__ rel_w,   // [32][7]
    float* __restrict__ out)
{
  const int idx = blockIdx.x * blockDim.x + threadIdx.x;
  const int w = idx & (HWDIM - 1);
  const int h = (idx >> 6) & (HWDIM - 1);
  const int o = (idx >> 12) & (COUT - 1);
  const int b = idx >> 18;

  const size_t plane = ((size_t)(b * COUT + o)) * NPIX;
  const float qv = q[plane + (h << 6) + w];

  float logit[KWIN * KWIN];
  float mx = -3.402823466e+38f;

#pragma unroll
  for (int dy = 0; dy < KWIN; ++dy) {
    const int y = h + dy - PADW;
    const bool yin = (unsigned)y < (unsigned)HWDIM;
#pragma unroll
    for (int dx = 0; dx < KWIN; ++dx) {
      const int xx = w + dx - PADW;
      const bool in = yin && ((unsigned)xx < (unsigned)HWDIM);
      float kv = 0.0f;
      if (in) kv = kf[plane + (y << 6) + xx];
      const float r = (o < 32) ? rel_h[o * KWIN + dy]
                               : rel_w[(o - 32) * KWIN + dx];
      const float l = qv * (kv + r);
      logit[dy * KWIN + dx] = l;
      mx = fmaxf(mx, l);
    }
  }

  float num = 0.0f, den = 0.0f;
#pragma unroll
  for (int dy = 0; dy < KWIN; ++dy) {
    const int y = h + dy - PADW;
    const bool yin = (unsigned)y < (unsigned)HWDIM;
#pragma unroll
    for (int dx = 0; dx < KWIN; ++dx) {
      const int xx = w + dx - PADW;
      const bool in = yin && ((unsigned)xx < (unsigned)HWDIM);
      const float e = __expf(logit[dy * KWIN + dx] - mx);
      float vv = 0.0f;
      if (in) vv = vf[plane + (y << 6) + xx];
      num += e * vv;
      den += e;
    }
  }
  out[idx] = num / den;
}

// ---------------------------------------------------------------------------
extern "C" void kernel_launch(void* const* d_in, const int* in_sizes, int n_in,
                              void* d_out, int out_size, void* d_ws, size_t ws_size,
                              hipStream_t stream) {
  const float* x     = (const float*)d_in[0];
  const float* wq    = (const float*)d_in[1];
  const float* wk    = (const float*)d_in[2];
  const float* wv    = (const float*)d_in[3];
  const float* rel_h = (const float*)d_in[4];
  const float* rel_w = (const float*)d_in[5];

  const int B = in_sizes[0] / (CIN * NPIX);   // 4

  float* qws = (float*)d_ws;
  float* kws = qws + (size_t)B * COUT * NPIX;
  float* vws = kws + (size_t)B * COUT * NPIX;

  // Stage 1: B*256 pixel-tiles, 8 waves (256 threads) per block.
  const int waves1  = B * 256;
  const int blocks1 = waves1 / 8;
  qkv_wmma_kernel<<<blocks1, 256, 0, stream>>>(x, wq, wk, wv, qws, kws, vws);

  // Stage 2: one thread per output element.
  const int total   = B * COUT * NPIX;
  attn_kernel<<<total / 256, 256, 0, stream>>>(qws, kws, vws, rel_h, rel_w,
                                               (float*)d_out);
}